// SEONN_Model_57758720197075
// MI455X (gfx1250) — compile-verified
//
#include <hip/hip_runtime.h>
#include <hip/hip_bf16.h>

#define N_NEURONS 16384
#define INPUT_SIZE 1024
#define OUTPUT_SIZE 1000
#define BATCH 32
#define PROP_STEPS 5

typedef __attribute__((ext_vector_type(2))) float v2f;
typedef __attribute__((ext_vector_type(8))) float v8f;

// Guaranteed-native, non-returning f32 atomic add (no CAS loop).
// WGP$ is read-only on CDNA5, so the RMW lands in the L2 atomic units
// (device-coherent); tracked by STOREcnt, never waited on per-op.
__device__ __forceinline__ void atomic_add_f32_native(float* p, float v) {
    asm volatile("global_atomic_add_f32 %0, %1, off"
                 :
                 : "v"((unsigned long long)(uintptr_t)p), "v"(v)
                 : "memory");
}

// ---------------------------------------------------------------------------
// state[n*32 + b] = x[b*1024 + n] for n < 1024, else 0; nstate = 0
// ---------------------------------------------------------------------------
__global__ void init_state_k(const float* __restrict__ x, float* __restrict__ state,
                             float* __restrict__ nstate, int total) {
    int t = blockIdx.x * blockDim.x + threadIdx.x;
    if (t < total) {
        int n = t >> 5;          // neuron
        int b = t & 31;          // batch (lane)
        state[t]  = (n < INPUT_SIZE) ? x[b * INPUT_SIZE + n] : 0.0f;
        nstate[t] = 0.0f;
    }
}

// ---------------------------------------------------------------------------
// Sparse message pass: one wave per edge (grid-stride). lane = batch index.
// Gather of state[col*32+lane] is one coalesced 128B line per edge; the 32
// lane atomics land on one 128B line of new_state. Edge scalars are wave-
// uniform loads. All edge data is L2-resident (32 MB << 192 MB L2).
// ---------------------------------------------------------------------------
__global__ void edge_spmm_k(const int* __restrict__ erow, const int* __restrict__ ecol,
                            const float* __restrict__ ew,
                            const float* __restrict__ state,
                            float* __restrict__ nstate, int nnz) {
    const int lane   = threadIdx.x & 31;
    const int wave   = (blockIdx.x * blockDim.x + threadIdx.x) >> 5;
    const int nwaves = (gridDim.x * blockDim.x) >> 5;
    for (int e = wave; e < nnz; e += nwaves) {
        int en = e + nwaves;                 // next iteration's edge
        if (en < nnz) {                      // gfx1250 global_prefetch_b8
            __builtin_prefetch(&erow[en], 0, 1);
            __builtin_prefetch(&ecol[en], 0, 1);
            __builtin_prefetch(&ew[en],   0, 1);
        }
        int   r = erow[e];
        int   c = ecol[e];
        float w = ew[e];
        float val = w * state[c * BATCH + lane];
        atomic_add_f32_native(&nstate[r * BATCH + lane], val);
    }
}

// ---------------------------------------------------------------------------
// state = gelu(state + new_state), exact-erf GELU (matches approximate=False).
// Also re-zeroes nstate for the next step (fuses the zero pass).
// ---------------------------------------------------------------------------
__global__ void update_gelu_k(float* __restrict__ state, float* __restrict__ nstate,
                              int total) {
    int t = blockIdx.x * blockDim.x + threadIdx.x;
    if (t < total) {
        float s = state[t] + nstate[t];
        nstate[t] = 0.0f;                    // ready for next step
        float g = 0.5f * s * (1.0f + erff(s * 0.70710678118654752440f));
        state[t] = g;
    }
}

// ---------------------------------------------------------------------------
// Output extraction via WMMA transpose on the FP32 matrix pipe.
// Per 16x16 tile: M[r][c] = state[(1024+j0+r)*32 + (b0+c)]  (r=neuron, c=batch)
// D = sum_{k=0..3} A_k x B_k, A_k = 16x4 identity slice (A_k[i][c]=d(i,4k+c)),
// B_k[c][j] = M[j][4k+c]  ==>  D[i][j] = M[j][i] = M^T.
// Fragment layouts (wave32, f32):
//   A 16x4:  lane L: row m = L&15; VGPR0 = K = 2*(L>>4), VGPR1 = K+1
//   B 4x16:  lane L: col j = L&15; VGPR0 = K = 2*(L>>4), VGPR1 = K+1
//   C/D 16x16: VGPR v, lane L: row i = v + 8*(L>>4), col j = L&15
// One 32-thread block (one wave, EXEC all ones) per tile.
// ---------------------------------------------------------------------------
__global__ void out_transpose_wmma_k(const float* __restrict__ state,
                                     float* __restrict__ out) {
    const int tile = blockIdx.x;         // 0..125
    const int jt = tile % 63;            // neuron tile (ceil(1000/16) = 63)
    const int bt = tile / 63;            // batch tile (0 or 1)
    const int j0 = jt * 16;
    const int b0 = bt * 16;
    const int L  = threadIdx.x;          // 0..31, all active
    const int j  = L & 15;
    const int h  = L >> 4;

    // lane L reads row (j0+j) of M: contiguous f32, coalesced within the wave
    const float* rowp = state + (size_t)(INPUT_SIZE + j0 + j) * BATCH + b0;

    v8f acc = {};
#pragma unroll
    for (int k = 0; k < 4; ++k) {
        const int kk = 4 * k + 2 * h;
        v2f bfrag;
        bfrag.x = rowp[kk + 0];
        bfrag.y = rowp[kk + 1];
        v2f afrag;
        afrag.x = (j == kk + 0) ? 1.0f : 0.0f;   // identity slice, row m = L&15
        afrag.y = (j == kk + 1) ? 1.0f : 0.0f;
        acc = __builtin_amdgcn_wmma_f32_16x16x4_f32(
            /*neg_a=*/false, afrag, /*neg_b=*/false, bfrag,
            /*c_mod=*/(short)0, acc, /*reuse_a=*/false, /*reuse_b=*/false);
    }

#pragma unroll
    for (int v = 0; v < 8; ++v) {
        int i  = v + 8 * h;              // batch within tile
        int jj = j0 + j;                 // output neuron index
        if (jj < OUTPUT_SIZE)
            out[(b0 + i) * OUTPUT_SIZE + jj] = acc[v];
    }
}

// ---------------------------------------------------------------------------
extern "C" void kernel_launch(void* const* d_in, const int* in_sizes, int n_in,
                              void* d_out, int out_size, void* d_ws, size_t ws_size,
                              hipStream_t stream) {
    const float* x    = (const float*)d_in[0];
    const int*   erow = (const int*)d_in[1];
    const int*   ecol = (const int*)d_in[2];
    const float* ew   = (const float*)d_in[3];
    const int    nnz  = in_sizes[1];

    float* out    = (float*)d_out;
    float* state  = (float*)d_ws;                               // 2 MB
    float* nstate = state + (size_t)N_NEURONS * BATCH;          // 2 MB

    const int total = N_NEURONS * BATCH;                        // 524288
    const int blk = 256;                                        // 8 waves/WG
    const int grd = (total + blk - 1) / blk;

    init_state_k<<<grd, blk, 0, stream>>>(x, state, nstate, total);

    for (int s = 0; s < PROP_STEPS; ++s) {
        edge_spmm_k<<<4096, blk, 0, stream>>>(erow, ecol, ew, state, nstate, nnz);
        update_gelu_k<<<grd, blk, 0, stream>>>(state, nstate, total);
    }

    // 126 tiles, one wave each
    out_transpose_wmma_k<<<126, 32, 0, stream>>>(state, out);
}